// RegularModel_90426241450317
// MI455X (gfx1250) — compile-verified
//
#include <hip/hip_runtime.h>
#include <math.h>

// MI455X / gfx1250, wave32. One workgroup (256 thr = 8 waves) per batch item.
// State (4096 complex) lives in LDS; gate layers applied as three 16x16
// complex matmuls via V_WMMA_F32_16X16X4_F32 (Kronecker-factored groups of
// 4 qubits); CNOT ladder applied as a composed linear permutation.

typedef __attribute__((ext_vector_type(2))) float v2f;
typedef __attribute__((ext_vector_type(8))) float v8f;

#define NQ      12
#define NL      4
#define QDIM    4096
#define THREADS 256

static_assert(QDIM == (1 << NQ), "dim mismatch");

struct c32 { float re, im; };
__device__ __forceinline__ c32 cmul(c32 a, c32 b) {
  return { a.re * b.re - a.im * b.im, a.re * b.im + a.im * b.re };
}
__device__ __forceinline__ c32 cadd(c32 a, c32 b) { return { a.re + b.re, a.im + b.im }; }

// r = p * q, 2x2 complex, row-major (i*2+j)
__device__ __forceinline__ void mat2mul(c32 r[4], const c32 p[4], const c32 q[4]) {
  r[0] = cadd(cmul(p[0], q[0]), cmul(p[1], q[2]));
  r[1] = cadd(cmul(p[0], q[1]), cmul(p[1], q[3]));
  r[2] = cadd(cmul(p[2], q[0]), cmul(p[3], q[2]));
  r[3] = cadd(cmul(p[2], q[1]), cmul(p[3], q[3]));
}

// D = A(16x16 complex) x B(16x16 complex tile of the state), in-place on the
// state LDS arrays. Tile element (k,n) of B lives at base + k*SK + n*SN.
// Each wave owns tiles {wave, wave+8}; footprints are disjoint per tile.
template <int SK, int SN, int BM>
__device__ __forceinline__ void apply_group(float* sre, float* sim,
                                            const float* AreG, const float* AimG,
                                            int wave, int half, int mn) {
#pragma unroll
  for (int tt = 0; tt < 2; ++tt) {
    const int tile = wave + tt * 8;
    const int base = tile * BM;
    v8f cre = {0.f, 0.f, 0.f, 0.f, 0.f, 0.f, 0.f, 0.f};
    v8f cim = {0.f, 0.f, 0.f, 0.f, 0.f, 0.f, 0.f, 0.f};
#pragma unroll
    for (int kc = 0; kc < 4; ++kc) {
      const int k0 = kc * 4 + 2 * half;           // this lane-half's K pair
      const int ar = mn * 16 + k0;                // A[m=mn][k0], A[m][k0+1]
      v2f Ar = { AreG[ar], AreG[ar + 1] };
      v2f Ai = { AimG[ar], AimG[ar + 1] };
      v2f An = -Ai;                                // f32 WMMA has no A-negate
      const int ib = base + k0 * SK + mn * SN;     // B[k0][n=mn], B[k0+1][n]
      v2f Br = { sre[ib], sre[ib + SK] };
      v2f Bi = { sim[ib], sim[ib + SK] };
      // Re += Ar*Br - Ai*Bi ; Im += Ar*Bi + Ai*Br
      cre = __builtin_amdgcn_wmma_f32_16x16x4_f32(false, Ar, false, Br, 0, cre, false, false);
      cre = __builtin_amdgcn_wmma_f32_16x16x4_f32(false, An, false, Bi, 0, cre, false, false);
      cim = __builtin_amdgcn_wmma_f32_16x16x4_f32(false, Ar, false, Bi, 0, cim, false, false);
      cim = __builtin_amdgcn_wmma_f32_16x16x4_f32(false, Ai, false, Br, 0, cim, false, false);
    }
    // C/D layout: vgpr v holds (M = v + 8*half, N = mn)
#pragma unroll
    for (int v = 0; v < 8; ++v) {
      const int m   = v + 8 * half;
      const int idx = base + m * SK + mn * SN;
      sre[idx] = cre[v];
      sim[idx] = cim[v];
    }
  }
}

__global__ __launch_bounds__(THREADS)
void qcircuit_parity_kernel(const float* __restrict__ x,
                            const float* __restrict__ thetas,
                            float* __restrict__ out) {
  __shared__ float sre[QDIM];
  __shared__ float sim[QDIM];
  __shared__ float Are[3 * 256];   // 16x16 group unitaries (real)
  __shared__ float Aim[3 * 256];   // 16x16 group unitaries (imag)
  __shared__ float gmre[NQ * 4];   // per-qubit composed 2x2 gates
  __shared__ float gmim[NQ * 4];
  __shared__ float red[THREADS];

  const int tid  = threadIdx.x;
  const int b    = blockIdx.x;
  const int lane = tid & 31;
  const int wave = tid >> 5;
  const int half = lane >> 4;
  const int mn   = lane & 15;

  // |0...0> initial state
  for (int j = tid; j < QDIM; j += THREADS) { sre[j] = 0.f; sim[j] = 0.f; }
  if (tid == 0) sre[0] = 1.f;

  // Feature-map angles for this batch element
  const float xv = x[b];
  const float x1 = asinf(xv);
  const float x2 = acosf(xv * xv);

  // Composed CNOT-ladder gather map: src(i) for amplitude gather.
  // state_out[i] = state_in[g01(g12(...g_{10,11}(i)...))], qubit j at bit 11-j.
  int csrc[16];
#pragma unroll
  for (int j = 0; j < 16; ++j) {
    int s = tid + j * THREADS;
#pragma unroll
    for (int jj = 10; jj >= 0; --jj)
      s ^= ((s >> (11 - jj)) & 1) << (10 - jj);
    csrc[j] = s;
  }

  __syncthreads();

  for (int l = 0; l < NL; ++l) {
    // ---- per-qubit composed 2x2 gate: RX(t2)*RZ(t1)*RX(t0)*RZ(x2)*RY(x1) ----
    if (tid < NQ) {
      const int q = tid;
      c32 M[4];
      { float s, c; __sincosf(0.5f * x1, &s, &c);          // RY(x1)
        M[0] = c32{c, 0.f}; M[1] = c32{-s, 0.f}; M[2] = c32{s, 0.f}; M[3] = c32{c, 0.f}; }
      { float s, c; __sincosf(0.5f * x2, &s, &c);          // RZ(x2)
        c32 G[4] = { c32{c, -s}, c32{0.f, 0.f}, c32{0.f, 0.f}, c32{c, s} };
        c32 T[4]; mat2mul(T, G, M);
        M[0] = T[0]; M[1] = T[1]; M[2] = T[2]; M[3] = T[3]; }
      const float t0 = thetas[(l * NQ + q) * 3 + 0];
      const float t1 = thetas[(l * NQ + q) * 3 + 1];
      const float t2 = thetas[(l * NQ + q) * 3 + 2];
      { float s, c; __sincosf(0.5f * t0, &s, &c);          // RX(t0)
        c32 G[4] = { c32{c, 0.f}, c32{0.f, -s}, c32{0.f, -s}, c32{c, 0.f} };
        c32 T[4]; mat2mul(T, G, M);
        M[0] = T[0]; M[1] = T[1]; M[2] = T[2]; M[3] = T[3]; }
      { float s, c; __sincosf(0.5f * t1, &s, &c);          // RZ(t1)
        c32 G[4] = { c32{c, -s}, c32{0.f, 0.f}, c32{0.f, 0.f}, c32{c, s} };
        c32 T[4]; mat2mul(T, G, M);
        M[0] = T[0]; M[1] = T[1]; M[2] = T[2]; M[3] = T[3]; }
      { float s, c; __sincosf(0.5f * t2, &s, &c);          // RX(t2)
        c32 G[4] = { c32{c, 0.f}, c32{0.f, -s}, c32{0.f, -s}, c32{c, 0.f} };
        c32 T[4]; mat2mul(T, G, M);
        M[0] = T[0]; M[1] = T[1]; M[2] = T[2]; M[3] = T[3]; }
#pragma unroll
      for (int k = 0; k < 4; ++k) { gmre[q * 4 + k] = M[k].re; gmim[q * 4 + k] = M[k].im; }
    }
    __syncthreads();

    // ---- 16x16 group unitaries via Kronecker product (1 entry/thread/group) ----
    {
      const int r = tid >> 4, cc = tid & 15;
#pragma unroll
      for (int g = 0; g < 3; ++g) {
        c32 acc = c32{1.f, 0.f};
#pragma unroll
        for (int j = 0; j < 4; ++j) {
          const int q  = 4 * g + j;
          const int bi = (r  >> (3 - j)) & 1;
          const int bj = (cc >> (3 - j)) & 1;
          acc = cmul(acc, c32{ gmre[q * 4 + bi * 2 + bj], gmim[q * 4 + bi * 2 + bj] });
        }
        Are[g * 256 + r * 16 + cc] = acc.re;
        Aim[g * 256 + r * 16 + cc] = acc.im;
      }
    }
    __syncthreads();

    // ---- apply the three group transforms with fp32 WMMA ----
    // state index i = a*256 + b*16 + c (qubits 0-3 -> a, 4-7 -> b, 8-11 -> c)
    apply_group<256, 1, 16 >(sre, sim, Are + 0,   Aim + 0,   wave, half, mn);  // over a
    __syncthreads();
    apply_group<16,  1, 256>(sre, sim, Are + 256, Aim + 256, wave, half, mn);  // over b
    __syncthreads();
    apply_group<1,  16, 256>(sre, sim, Are + 512, Aim + 512, wave, half, mn);  // over c
    __syncthreads();

    // ---- CNOT ladder: composed linear permutation, register-buffered gather ----
    float tr[16], ti[16];
#pragma unroll
    for (int j = 0; j < 16; ++j) { tr[j] = sre[csrc[j]]; ti[j] = sim[csrc[j]]; }
    __syncthreads();
#pragma unroll
    for (int j = 0; j < 16; ++j) {
      const int i = tid + j * THREADS;
      sre[i] = tr[j]; sim[i] = ti[j];
    }
    __syncthreads();
  }

  // ---- parity expectation <Z...Z> = sum |amp|^2 * (-1)^popcount ----
  float acc = 0.f;
  for (int j = tid; j < QDIM; j += THREADS) {
    const float p = (__popc(j) & 1) ? -1.f : 1.f;
    acc = fmaf(sre[j] * sre[j] + sim[j] * sim[j], p, acc);
  }
  red[tid] = acc;
  __syncthreads();
  for (int off = THREADS / 2; off > 0; off >>= 1) {
    if (tid < off) red[tid] += red[tid + off];
    __syncthreads();
  }
  if (tid == 0) out[b] = red[0];
}

extern "C" void kernel_launch(void* const* d_in, const int* in_sizes, int n_in,
                              void* d_out, int out_size, void* d_ws, size_t ws_size,
                              hipStream_t stream) {
  (void)n_in; (void)d_ws; (void)ws_size; (void)out_size;
  const float* x      = (const float*)d_in[0];
  const float* thetas = (const float*)d_in[1];
  float* out          = (float*)d_out;
  const int batch     = in_sizes[0];  // 4096
  qcircuit_parity_kernel<<<dim3(batch), dim3(THREADS), 0, stream>>>(x, thetas, out);
}